// SelfAttention_16045997818295
// MI455X (gfx1250) — compile-verified
//
#include <hip/hip_runtime.h>
#include <math.h>

typedef __attribute__((ext_vector_type(2))) float v2f;
typedef __attribute__((ext_vector_type(4))) float v4f;
typedef __attribute__((ext_vector_type(8))) float v8f;

#define SEQ   2048
#define DKD   64
#define QT    32            // query rows per block
#define SP    2052          // padded LDS row stride (floats) -> bank-conflict free
#define NBH   64            // B*H
#define NQT   (SEQ / QT)    // 64 q-tiles per head

__global__ __launch_bounds__(256) void attn_fwd_f32wmma(
    const float* __restrict__ Q, const float* __restrict__ K,
    const float* __restrict__ V, float* __restrict__ Out,
    float* __restrict__ Attn)
{
    extern __shared__ float smem[];
    float* sc   = smem;               // [QT][SP] scores -> probs
    float* red  = smem + QT * SP;     // [QT][8] partial reductions
    float* rowm = red + QT * 8;       // [QT] row max
    float* rinv = rowm + QT;          // [QT] 1/row_sum

    const int tid  = threadIdx.x;
    const int lane = tid & 31;
    const int w    = tid >> 5;                 // wave 0..7
    const int qt   = blockIdx.x % NQT;
    const int bh   = blockIdx.x / NQT;
    const int qbase = qt * QT;

    const float* Qh = Q + (size_t)bh * SEQ * DKD;
    const float* Kh = K + (size_t)bh * SEQ * DKD;
    const float* Vh = V + (size_t)bh * SEQ * DKD;
    float* outh  = Out  + (size_t)bh * SEQ * DKD;
    float* attnh = Attn + (size_t)bh * SEQ * SEQ;

    const int m16   = lane & 15;               // row (A) / col (B,C,D) within tile
    const int hhalf = lane >> 4;               // 0 or 1
    const int koff  = hhalf * 2;               // k sub-offset for A/B fragments
    const float scale = 0.125f;                // 1/sqrt(64)

    // ---------------- Phase 1: S = scale * Q K^T (causal) via WMMA ----------------
    const int qs     = qbase + (w & 1) * 16;   // this wave's 16-row q block (global row)
    const int kstrip = w >> 1;                 // 0..3 : k-tile strip

    // Preload Q fragments for all 16 k-steps (k = 0..63)
    v2f aQ[16];
#pragma unroll
    for (int kk = 0; kk < 16; ++kk)
        aQ[kk] = *(const v2f*)(Qh + (size_t)(qs + m16) * DKD + 4 * kk + koff);

    const int ktmax = qs >> 4;                 // last 16-wide k tile needed (causal)
    for (int kt = kstrip; kt <= ktmax; kt += 4) {
        const int kbase = kt * 16;
        v8f acc = {};
#pragma unroll
        for (int kk = 0; kk < 16; ++kk) {
            v2f bK = *(const v2f*)(Kh + (size_t)(kbase + m16) * DKD + 4 * kk + koff);
            acc = __builtin_amdgcn_wmma_f32_16x16x4_f32(
                      false, aQ[kk], false, bK, (short)0, acc, false, false);
        }
#pragma unroll
        for (int v = 0; v < 8; ++v) {
            const int gq = qs + v + 8 * hhalf;     // global query row
            const int gk = kbase + m16;            // global key col
            float s = acc[v] * scale;
            if (gk > gq) s = -INFINITY;            // causal mask
            sc[(gq - qbase) * SP + gk] = s;
        }
    }
    __syncthreads();

    // ---------------- Phase 2: row max and sum(exp(s-m)) ----------------
    {
        const int r  = tid >> 3;                   // 0..31
        const int j  = tid & 7;                    // 0..7
        const int gq = qbase + r;
        const int wlimit = (gq & ~15) + 16;        // written region bound (mult of 16)
        const float* row = sc + r * SP;
        const int k0   = j * 256;
        const int kend = (wlimit < k0 + 256) ? wlimit : (k0 + 256);

        float mloc = -INFINITY;
        for (int k = k0; k < kend; k += 4) {
            v4f s4 = *(const v4f*)(row + k);
            mloc = fmaxf(mloc, fmaxf(fmaxf(s4.x, s4.y), fmaxf(s4.z, s4.w)));
        }
        red[r * 8 + j] = mloc;
        __syncthreads();
        if (j == 0) {
            float m = red[r * 8];
            for (int t = 1; t < 8; ++t) m = fmaxf(m, red[r * 8 + t]);
            rowm[r] = m;
        }
        __syncthreads();
        const float m = rowm[r];
        float lsum = 0.f;
        for (int k = k0; k < kend; k += 4) {
            v4f s4 = *(const v4f*)(row + k);
            lsum += __expf(s4.x - m) + __expf(s4.y - m)
                  + __expf(s4.z - m) + __expf(s4.w - m);
        }
        red[r * 8 + j] = lsum;
        __syncthreads();
        if (j == 0) {
            float l = 0.f;
            for (int t = 0; t < 8; ++t) l += red[r * 8 + t];
            rinv[r] = 1.0f / l;
        }
        __syncthreads();
    }

    // ---------------- Phase 3: P = softmax(S); write attn (NT), keep P in LDS ----
    {
        float* attnq = attnh + (size_t)qbase * SEQ;
        for (int it = 0; it < (QT * SEQ) / (256 * 4); ++it) {   // 64 iterations
            const int idx = (it * 256 + tid) * 4;               // element idx in 32x2048
            const int r   = idx >> 11;                          // row 0..31
            const int k   = idx & (SEQ - 1);
            const int gq  = qbase + r;
            const int wlimit = (gq & ~15) + 16;
            v4f p;
            if (k < wlimit) {                                   // whole v4f inside
                const float m  = rowm[r];
                const float ri = rinv[r];
                v4f s4 = *(const v4f*)(sc + r * SP + k);
                p.x = __expf(s4.x - m) * ri;                    // -inf -> 0
                p.y = __expf(s4.y - m) * ri;
                p.z = __expf(s4.z - m) * ri;
                p.w = __expf(s4.w - m) * ri;
                *(v4f*)(sc + r * SP + k) = p;
            } else {
                p = (v4f){0.f, 0.f, 0.f, 0.f};
            }
            __builtin_nontemporal_store(p, (v4f*)(attnq + (size_t)idx));
        }
    }
    __syncthreads();

    // ---------------- Phase 4: Out = P @ V via WMMA ----------------
    // nkb = (qs+16)/4 is always a multiple of 4 (qs % 16 == 0), so we can
    // hand-unroll x4 with no remainder: batch 4 ds_load_b64 + 8 global loads
    // ahead of 4 chained WMMAs for load/compute overlap.
    {
        const int dbase = (w >> 1) * 16;          // 4 strips of 16 cols cover DK=64
        const int rr    = qs - qbase;             // 0 or 16 (LDS row base)
        const int nkb   = (qs + 16) >> 2;         // multiple of 4
        const float* prow = sc + (rr + m16) * SP + koff;
        const float* vcol = Vh + dbase + m16 + (size_t)koff * DKD;
        v8f acc = {};
        for (int kb = 0; kb < nkb; kb += 4) {
            v2f aP0 = *(const v2f*)(prow + 4 * kb);
            v2f aP1 = *(const v2f*)(prow + 4 * kb + 4);
            v2f aP2 = *(const v2f*)(prow + 4 * kb + 8);
            v2f aP3 = *(const v2f*)(prow + 4 * kb + 12);
            const float* vb = vcol + (size_t)(4 * kb) * DKD;
            v2f bV0, bV1, bV2, bV3;
            bV0.x = vb[0];            bV0.y = vb[DKD];
            bV1.x = vb[4 * DKD];      bV1.y = vb[5 * DKD];
            bV2.x = vb[8 * DKD];      bV2.y = vb[9 * DKD];
            bV3.x = vb[12 * DKD];     bV3.y = vb[13 * DKD];
            acc = __builtin_amdgcn_wmma_f32_16x16x4_f32(
                      false, aP0, false, bV0, (short)0, acc, false, false);
            acc = __builtin_amdgcn_wmma_f32_16x16x4_f32(
                      false, aP1, false, bV1, (short)0, acc, false, false);
            acc = __builtin_amdgcn_wmma_f32_16x16x4_f32(
                      false, aP2, false, bV2, (short)0, acc, false, false);
            acc = __builtin_amdgcn_wmma_f32_16x16x4_f32(
                      false, aP3, false, bV3, (short)0, acc, false, false);
        }
#pragma unroll
        for (int v = 0; v < 8; ++v) {
            const int gq = qs + v + 8 * hhalf;
            outh[(size_t)gq * DKD + dbase + m16] = acc[v];
        }
    }
}

extern "C" void kernel_launch(void* const* d_in, const int* in_sizes, int n_in,
                              void* d_out, int out_size, void* d_ws, size_t ws_size,
                              hipStream_t stream) {
    (void)in_sizes; (void)n_in; (void)out_size; (void)d_ws; (void)ws_size;
    const float* Q = (const float*)d_in[0];
    const float* K = (const float*)d_in[1];
    const float* V = (const float*)d_in[2];
    // d_in[3] is the causal mask; causality is applied analytically in-kernel.

    float* Out  = (float*)d_out;                                   // [B,H,S,DK]
    float* Attn = Out + (size_t)4 * 16 * SEQ * DKD;                // [B,H,S,S]

    const size_t smem_bytes = (size_t)(QT * SP + QT * 8 + QT + QT) * sizeof(float);
    static_assert(QT * SP * 4 < 320 * 1024, "fits WGP LDS");

    // Allow >default dynamic LDS (258 KB of the 320 KB WGP pool).
    (void)hipFuncSetAttribute((const void*)attn_fwd_f32wmma,
                              hipFuncAttributeMaxDynamicSharedMemorySize,
                              (int)smem_bytes);

    dim3 grid(NBH * NQT);   // 64 heads * 64 q-tiles = 4096
    dim3 block(256);
    attn_fwd_f32wmma<<<grid, block, smem_bytes, stream>>>(Q, K, V, Out, Attn);
}